// SkillModule_52432960750160
// MI455X (gfx1250) — compile-verified
//
#include <hip/hip_runtime.h>
#include <hip/hip_bf16.h>

#define N_TOK   32768
#define D_EMB   128
#define K_CODE  2048
#define DECAY   0.99f
#define EPS_VQ  1e-5f

typedef __attribute__((ext_vector_type(2))) float        v2f;
typedef __attribute__((ext_vector_type(8))) float        v8f;
typedef __attribute__((ext_vector_type(4))) unsigned int v4u;
typedef __attribute__((ext_vector_type(8))) int          v8i;
typedef __attribute__((ext_vector_type(4))) int          v4i;

// ---------------------------------------------------------------------------
// K0: wsq[k] = sum_d weight[d][k]^2 ; zero counts & n ; seed embed_avg_new.
// ---------------------------------------------------------------------------
__global__ void k0_init(const float* __restrict__ weight,
                        const float* __restrict__ embed_avg,
                        float* __restrict__ out_eavg_new,
                        float* __restrict__ wsq,
                        float* __restrict__ counts,
                        float* __restrict__ nsum) {
    int i = blockIdx.x * blockDim.x + threadIdx.x;
    if (i < D_EMB * K_CODE) out_eavg_new[i] = embed_avg[i] * DECAY;
    if (i < K_CODE) {
        float s = 0.f;
#pragma unroll 8
        for (int d = 0; d < D_EMB; ++d) {
            float w = weight[(size_t)d * K_CODE + i];
            s += w * w;
        }
        wsq[i] = s;
        counts[i] = 0.f;
    }
    if (i == 0) *nsum = 0.f;
}

// ---------------------------------------------------------------------------
// TDM: issue an async DMA of weight[0:128][colBase:colBase+16] into LDS.
// D# per CDNA5 ISA ch.8: 2-D tile, data_size=4B, tensor_dim0_stride=K.
// Wave-level op (EXEC ignored); tracked with TENSORcnt.
// This toolchain (clang-23 / therock-10.0) uses the 6-arg builtin form:
//   (uint32x4 g0, int32x8 g1, int32x4, int32x4, int32x8, i32 cpol)
// ---------------------------------------------------------------------------
__device__ __forceinline__ void tdm_load_tile(const float* weight, int colBase,
                                              unsigned int ldsByteAddr) {
    unsigned long long ga =
        (unsigned long long)(const void*)weight + (unsigned long long)colBase * 4ull;
    v4u g0;
    g0[0] = 1u;                                        // count=1, user D#
    g0[1] = ldsByteAddr;                               // lds_addr
    g0[2] = (unsigned int)(ga & 0xFFFFFFFFull);        // global_addr[31:0]
    g0[3] = (unsigned int)((ga >> 32) & 0x1FFFFFFull)  // global_addr[56:32]
          | (2u << 30);                                // type = 2 ("image")
    v8i g1;
    g1[0] = (int)(2u << 16);            // workgroup_mask=0 | data_size=2 (4B)
    g1[1] = (int)(2048u << 16);         // tensor_dim0[15:0] = K = 2048
    g1[2] = (int)(128u << 16);          // tensor_dim0[31:16]=0 | tensor_dim1[15:0]=128
    g1[3] = (int)(16u << 16);           // tensor_dim1[31:16]=0 | tile_dim0 = 16
    g1[4] = 128;                        // tile_dim1 = 128 | tile_dim2 = 0
    g1[5] = 2048;                       // tensor_dim0_stride[31:0] = K
    g1[6] = 0;                          // stride[47:32] | tensor_dim1_stride lo
    g1[7] = 0;
    v4i g2 = {0, 0, 0, 0};
    v4i g3 = {0, 0, 0, 0};
    v8i g4 = {0, 0, 0, 0, 0, 0, 0, 0};
    __builtin_amdgcn_tensor_load_to_lds(g0, g1, g2, g3, g4, 0);
}

// ---------------------------------------------------------------------------
// K1: fused distance GEMM (v_wmma_f32_16x16x4_f32) + per-row argmin.
// Block = 128 threads = 4 waves; each wave owns 16 rows. Weight tiles are
// streamed into a double-buffered LDS tile by the Tensor Data Mover so the
// fetch of tile kt+1 overlaps the 32 WMMAs of tile kt.
// score(n,k) = wsq[k] - 2 * (x_n . w_k)   (same argmin as full L2 distance)
// ---------------------------------------------------------------------------
__global__ void __launch_bounds__(128) k1_wmma_argmin(
        const float* __restrict__ x,
        const float* __restrict__ weight,
        const float* __restrict__ wsq,
        int* __restrict__ argmin_ws) {
    __shared__ float ldsW[2][D_EMB * 16];   // double buffer, 2 x 8 KB

    const int tid  = threadIdx.x;
    const int wave = tid >> 5;
    const int lane = tid & 31;
    const int m    = lane & 15;          // row (A) / col (B) within tile
    const int half = lane >> 4;          // K-pair selector
    const int rowBase = (blockIdx.x * 4 + wave) * 16;
    const int NT = K_CODE / 16;

    const unsigned int ldsAddr[2] = {
        (unsigned int)(size_t)(&ldsW[0][0]),
        (unsigned int)(size_t)(&ldsW[1][0])
    };

    // A fragments: 16x4 f32 per chunk, 32 chunks cover D=128. Lane m holds
    // row (rowBase+m); half 0 -> K=d,d+1 ; half 1 -> K=d+2,d+3.
    v2f a[32];
    const float* xrow = x + (size_t)(rowBase + m) * D_EMB;
#pragma unroll
    for (int c = 0; c < 32; ++c)
        a[c] = *(const v2f*)(xrow + 4 * c + 2 * half);

    // Per-lane running argmin over the columns this lane covers (col%16==m).
    float bestS[8];
    int   bestI[8];
#pragma unroll
    for (int i = 0; i < 8; ++i) { bestS[i] = 3.4e38f; bestI[i] = 0; }

    // Prologue: DMA tile 0 into buffer 0 (wave 0 issues; EXEC-independent).
    if (tid < 32) tdm_load_tile(weight, 0, ldsAddr[0]);

    for (int kt = 0; kt < NT; ++kt) {
        const int par = kt & 1;

        if (tid < 32) {
            if (kt + 1 < NT) {
                tdm_load_tile(weight, (kt + 1) * 16, ldsAddr[par ^ 1]);
                __builtin_amdgcn_s_wait_tensorcnt(1);   // tile kt landed
            } else {
                __builtin_amdgcn_s_wait_tensorcnt(0);   // last tile landed
            }
        }
        __syncthreads();   // publish buffer `par` to all 4 waves

        const float* wt = &ldsW[par][0];
        const int colBase = kt * 16;
        const float sq = wsq[colBase + m];

        v8f acc0 = {};
        v8f acc1 = {};
#pragma unroll
        for (int c = 0; c < 32; c += 2) {
            // B fragment (4x16 f32): lane m holds col (colBase+m);
            // half 0 -> K=d,d+1 rows ; half 1 -> K=d+2,d+3 rows.
            v2f b0, b1;
            b0[0] = wt[(4 * c + 2 * half + 0) * 16 + m];
            b0[1] = wt[(4 * c + 2 * half + 1) * 16 + m];
            b1[0] = wt[(4 * (c + 1) + 2 * half + 0) * 16 + m];
            b1[1] = wt[(4 * (c + 1) + 2 * half + 1) * 16 + m];
            acc0 = __builtin_amdgcn_wmma_f32_16x16x4_f32(
                false, a[c], false, b0, (short)0, acc0, false, false);
            acc1 = __builtin_amdgcn_wmma_f32_16x16x4_f32(
                false, a[c + 1], false, b1, (short)0, acc1, false, false);
        }
        v8f acc = acc0 + acc1;

        // C/D layout: VGPR i, half h holds row (i + 8h), col = colBase + m.
#pragma unroll
        for (int i = 0; i < 8; ++i) {
            float s = sq - 2.0f * acc[i];
            if (s < bestS[i]) { bestS[i] = s; bestI[i] = colBase + m; }
        }
        __syncthreads();   // all waves done reading buffer `par`
    }

    // One 16-lane butterfly per row at the end (first-minimum tie-break).
#pragma unroll
    for (int i = 0; i < 8; ++i) {
        float s   = bestS[i];
        int   idx = bestI[i];
#pragma unroll
        for (int delta = 1; delta < 16; delta <<= 1) {
            float so = __shfl_xor(s, delta, 32);
            int   io = __shfl_xor(idx, delta, 32);
            if (so < s || (so == s && io < idx)) { s = so; idx = io; }
        }
        bestI[i] = idx;
    }

    // Lane 0 holds rows rowBase+0..7, lane 16 holds rows rowBase+8..15.
    if (m == 0) {
#pragma unroll
        for (int i = 0; i < 8; ++i)
            argmin_ws[rowBase + 8 * half + i] = bestI[i];
    }
}

// ---------------------------------------------------------------------------
// K2: per-row codebook gather + EMA scatter. One 128-thread block per row.
// ---------------------------------------------------------------------------
__global__ void __launch_bounds__(128) k2_gather_scatter(
        const float* __restrict__ x,
        const float* __restrict__ weight,
        const int* __restrict__ argmin_ws,
        float* __restrict__ out_result,
        float* __restrict__ out_argmin_f,
        float* __restrict__ out_eavg_new,
        float* __restrict__ counts) {
    const int row  = blockIdx.x;
    const int d    = threadIdx.x;
    const int code = argmin_ws[row];

    // result = x + stopgrad(quantized - x) == quantized numerically
    out_result[(size_t)row * D_EMB + d] = weight[(size_t)d * K_CODE + code];

    // embed_avg_new already seeded with embed_avg*DECAY; accumulate segment sum
    atomicAdd(&out_eavg_new[(size_t)d * K_CODE + code],
              x[(size_t)row * D_EMB + d] * (1.0f - DECAY));

    if (d == 0) {
        out_argmin_f[row] = (float)code;
        atomicAdd(&counts[code], 1.0f);
    }
}

// ---------------------------------------------------------------------------
// K3: cluster_size EMA with zero-clamp; accumulate n = sum(cluster_size_new).
// ---------------------------------------------------------------------------
__global__ void k3_cluster(const float* __restrict__ cluster_size,
                           const float* __restrict__ counts,
                           float* __restrict__ out_cs_new,
                           float* __restrict__ nsum) {
    int k = blockIdx.x * blockDim.x + threadIdx.x;
    if (k >= K_CODE) return;
    float c = counts[k];
    float clamped = (c == 0.f) ? 1.f : c;
    float csn = cluster_size[k] * DECAY + clamped * (1.f - DECAY);
    out_cs_new[k] = csn;
    atomicAdd(nsum, csn);
}

// ---------------------------------------------------------------------------
// K4: Laplace smoothing + normalized codebook.
// ---------------------------------------------------------------------------
__global__ void k4_weight(const float* __restrict__ out_eavg_new,
                          const float* __restrict__ out_cs_new,
                          const float* __restrict__ nsum,
                          float* __restrict__ out_weight_new) {
    int i = blockIdx.x * blockDim.x + threadIdx.x;
    if (i >= D_EMB * K_CODE) return;
    int k = i & (K_CODE - 1);
    float n  = *nsum;
    float cs = (out_cs_new[k] + EPS_VQ) / (n + K_CODE * EPS_VQ) * n;
    out_weight_new[i] = out_eavg_new[i] / cs;
}

// ---------------------------------------------------------------------------
extern "C" void kernel_launch(void* const* d_in, const int* in_sizes, int n_in,
                              void* d_out, int out_size, void* d_ws, size_t ws_size,
                              hipStream_t stream) {
    const float* x            = (const float*)d_in[0];   // (N, D)
    const float* weight       = (const float*)d_in[1];   // (D, K)
    const float* cluster_size = (const float*)d_in[2];   // (K,)
    const float* embed_avg    = (const float*)d_in[3];   // (D, K)

    // Output tuple, flat concat: result, argmin, weight_new, cluster_size_new, embed_avg_new
    float* out        = (float*)d_out;
    float* out_result = out;                                   // N*D = 4194304
    float* out_argmin = out_result + (size_t)N_TOK * D_EMB;    // N   = 32768
    float* out_weight = out_argmin + N_TOK;                    // D*K = 262144
    float* out_cs     = out_weight + (size_t)D_EMB * K_CODE;   // K   = 2048
    float* out_eavg   = out_cs + K_CODE;                       // D*K = 262144

    // Workspace: wsq(K) | counts(K) | nsum(pad 64) | argmin(N) ~= 148 KB
    float* wsq       = (float*)d_ws;
    float* counts    = wsq + K_CODE;
    float* nsum      = counts + K_CODE;
    int*   argmin_ws = (int*)(nsum + 64);

    k0_init<<<(D_EMB * K_CODE + 255) / 256, 256, 0, stream>>>(
        weight, embed_avg, out_eavg, wsq, counts, nsum);

    k1_wmma_argmin<<<N_TOK / 64, 128, 0, stream>>>(x, weight, wsq, argmin_ws);

    k2_gather_scatter<<<N_TOK, 128, 0, stream>>>(
        x, weight, argmin_ws, out_result, out_argmin, out_eavg, counts);

    k3_cluster<<<(K_CODE + 255) / 256, 256, 0, stream>>>(
        cluster_size, counts, out_cs, nsum);

    k4_weight<<<(D_EMB * K_CODE + 255) / 256, 256, 0, stream>>>(
        out_eavg, out_cs, nsum, out_weight);
}